// EdgeEncoding_2216203124823
// MI455X (gfx1250) — compile-verified
//
#include <hip/hip_runtime.h>

// EdgeEncoding for MI455X (gfx1250, wave32).
//
// Roofline: ~160 MFLOP vs ~120+ MB of real memory traffic at 23.3 TB/s
// -> pure bandwidth problem (~5us floor). Compute is free, so the per-pair
// 80-long dot product is folded into chained V_WMMA_F32_16X16X4_F32 ops:
//   A (16x4): row M=0 carries the flattened edge_vector chunk (other rows 0)
//   B (4x16): column n carries pair n's gathered/masked feature chunk
//   D[0][n]  = accumulated dot for pair n  (lands in acc[0] on lane n)
// One wave handles 16 pairs; 20 K=4 chunks cover the 5x16 flattened dims.

typedef __attribute__((ext_vector_type(2))) float v2f;
typedef __attribute__((ext_vector_type(8))) float v8f;

#define N_NODES  4096
#define N_EDGES  100000
#define N_PAIRS  1000000
#define MAX_PATH 5
#define EDGE_DIM 16

__global__ void ee_zero_kernel(float4* __restrict__ out, int n4) {
    int i = blockIdx.x * blockDim.x + threadIdx.x;
    if (i < n4) out[i] = make_float4(0.f, 0.f, 0.f, 0.f);
}

__global__ __launch_bounds__(128) void ee_wmma_kernel(
    const float* __restrict__ edge_attr,    // [N_EDGES, 16]
    const float* __restrict__ edge_vector,  // [MAX_PATH, 16]
    const int*   __restrict__ src_idx,      // [N_PAIRS]
    const int*   __restrict__ dst_idx,      // [N_PAIRS]
    const int*   __restrict__ path_idx,     // [N_PAIRS, MAX_PATH], -1 = pad
    float*       __restrict__ out)          // [N_NODES, N_NODES]
{
    const int lane = threadIdx.x & 31;
    const int wid  = threadIdx.x >> 5;
    const int m    = lane & 15;   // pair slot within the 16-pair group
    const int h    = lane >> 4;   // K-half: lanes 0-15 -> K{0,1}, 16-31 -> K{2,3}
    const int pair = (blockIdx.x * 4 + wid) * 16 + m;   // exact cover, no tail

    v8f acc = {};   // 16x16 f32 C/D accumulator; we consume row M=0 only
    int cnt = 0;

    for (int l = 0; l < MAX_PATH; ++l) {
        // lanes m and m+16 read the same index -> merged by coalescer
        const int  e     = path_idx[pair * MAX_PATH + l];
        const bool valid = (e >= 0);
        cnt += valid ? 1 : 0;

        const float* row  = edge_attr + (long)(valid ? e : 0) * EDGE_DIM;
        const float* wrow = edge_vector + l * EDGE_DIM;

#pragma unroll
        for (int t = 0; t < 4; ++t) {
            const int d0 = 4 * t + 2 * h;   // this lane's two dims of the K=4 chunk

            // B (4x16): VGPR0 = K{0|2} by lane half, VGPR1 = K{1|3}.
            // Column n = pair n's (masked) features -> 8B aligned float2 gather.
            v2f fv = *(const v2f*)(row + d0);
            v2f b;
            b.x = valid ? fv.x : 0.f;
            b.y = valid ? fv.y : 0.f;

            // A (16x4): only row M=0 nonzero, holding the edge_vector chunk.
            // Lane 0 -> (w[4t], w[4t+1]); lane 16 -> (w[4t+2], w[4t+3]).
            v2f wv = *(const v2f*)(wrow + d0);   // 2 distinct addrs/wave, cache-hot
            v2f a;
            a.x = (m == 0) ? wv.x : 0.f;
            a.y = (m == 0) ? wv.y : 0.f;

            // (neg_a, A, neg_b, B, c_mod, C, reuse_a, reuse_b)
            acc = __builtin_amdgcn_wmma_f32_16x16x4_f32(
                false, a, false, b, (short)0, acc, false, false);
        }
    }

    // D[0][n] lives in component 0 of lanes 0..15 (N = lane).
    if (h == 0) {
        float v = (cnt > 0) ? (acc[0] / (float)cnt) : 0.f;
        long  o = (long)src_idx[pair] * N_NODES + (long)dst_idx[pair];
        out[o] = v;
    }
}

extern "C" void kernel_launch(void* const* d_in, const int* in_sizes, int n_in,
                              void* d_out, int out_size, void* d_ws, size_t ws_size,
                              hipStream_t stream) {
    // setup_inputs order: x, edge_attr, edge_vector, src_idx, dst_idx, path_idx
    const float* edge_attr   = (const float*)d_in[1];
    const float* edge_vector = (const float*)d_in[2];
    const int*   src_idx     = (const int*)d_in[3];
    const int*   dst_idx     = (const int*)d_in[4];
    const int*   path_idx    = (const int*)d_in[5];
    float*       out         = (float*)d_out;

    // 1) zero the 4096x4096 output (64 MB) with 16B stores
    int n4 = out_size / 4;   // out_size = 16,777,216, divisible by 4
    ee_zero_kernel<<<(n4 + 255) / 256, 256, 0, stream>>>((float4*)out, n4);

    // 2) gather + WMMA dot + scatter; 4 waves/block * 16 pairs/wave
    ee_wmma_kernel<<<N_PAIRS / 64, 128, 0, stream>>>(
        edge_attr, edge_vector, src_idx, dst_idx, path_idx, out);
}